// FPS_74457553043647
// MI455X (gfx1250) — compile-verified
//
#include <hip/hip_runtime.h>
#include <stdint.h>

// Problem constants (from reference): N=400000, WIDTH=64, STRIDE=400 -> m=1000
#define NPTS   400000
#define FWIDTH 64
#define MSAMP  (NPTS / 400)   // 1000
#define NBLK   128
#define TPB    256
#define RPB    (NPTS / NBLK)  // 3125 points per block, exact

typedef unsigned int v4u __attribute__((ext_vector_type(4)));
typedef int          v8i __attribute__((ext_vector_type(8)));
typedef int          v4i __attribute__((ext_vector_type(4)));

// Pack (value, index) into u64 so that u64-max == argmax with smallest-index
// tie-break (distances are always >= 0, so f32 bit pattern is order-preserving).
__device__ __forceinline__ unsigned long long pack_vi(float v, unsigned idx) {
    return ((unsigned long long)__float_as_uint(v) << 32) |
           (unsigned long long)(0xFFFFFFFFu - idx);
}

__device__ __forceinline__ unsigned long long u64max(unsigned long long a,
                                                     unsigned long long b) {
    return a > b ? a : b;
}

__global__ void fps_init_kernel(unsigned int* counter, int* out) {
    if (threadIdx.x == 0 && blockIdx.x == 0) {
        *counter = 0u;   // monotonic arrival counter, reset every launch
        out[0]   = 0;    // FPS seeds with index 0
    }
}

__global__ __launch_bounds__(TPB, 1) void fps_main_kernel(
        const float* __restrict__ feats,          // [NPTS, 64] f32
        int* __restrict__ out,                    // [MSAMP] int32 indices
        unsigned long long* __restrict__ blk_best,// [NBLK] per-block packed best
        unsigned int* __restrict__ counter)       // arrival counter
{
    // Block-local slice lives in LDS for the whole kernel:
    // tile[i] = (x, y, z, running_min_dist)
    __shared__ float4 tile[RPB];                          // 50,000 B
    __shared__ unsigned long long wred[TPB / 32];         // wave partials
    __shared__ float qshared[3];                          // current query point

    const int tid  = threadIdx.x;
    const int b    = blockIdx.x;
    const int base = b * RPB;

    // ---- Stage this block's 3125 rows (first 4 floats of each 64-float row)
    // ---- into LDS via the Tensor Data Mover (one 2-D tiled DMA per block).
#if __has_builtin(__builtin_amdgcn_tensor_load_to_lds)
    if (tid < 32) {  // wave 0 issues the TDM op (EXEC ignored by TDM)
        // Low 32 bits of a flat shared-aperture address are the LDS byte
        // offset on gfx1250 (ISA 10.2 aperture mapping).
        unsigned lds_addr = (unsigned)(uintptr_t)(&tile[0]);
        unsigned long long ga =
            (unsigned long long)(uintptr_t)(feats + (size_t)base * FWIDTH);

        v4u g0;
        g0.x = 0x1u;                                     // count=1, user mode
        g0.y = lds_addr;                                 // lds_addr [63:32]
        g0.z = (unsigned)(ga & 0xFFFFFFFFull);           // global_addr[31:0]
        g0.w = (unsigned)((ga >> 32) & 0x01FFFFFFull)    // global_addr[56:32]
             | (2u << 30);                               // type=2 ("image")

        v8i g1;
        g1[0] = 0x00020000;                   // wg_mask=0, data_size=2 (4 B)
        g1[1] = (int)(((unsigned)FWIDTH & 0xFFFFu) << 16); // tensor_dim0 lo16
        g1[2] = (int)(((unsigned)RPB & 0xFFFFu) << 16);    // dim0 hi=0 | tensor_dim1 lo16
        g1[3] = (int)(4u << 16);              // tensor_dim1 hi=0 | tile_dim0=4
        g1[4] = (int)(RPB & 0xFFFF);          // tile_dim1=3125, tile_dim2=0
        g1[5] = FWIDTH;                       // tensor_dim0_stride = 64 elems
        g1[6] = 0;                            // stride hi | dim1_stride lo
        g1[7] = 0;                            // dim1_stride hi (unused, 2-D)

        v4i gz4 = {0, 0, 0, 0};               // groups 2/3 unused (2-D tensor)
        v8i gz8 = {0, 0, 0, 0, 0, 0, 0, 0};   // extra group (6-arg toolchain)
        __builtin_amdgcn_tensor_load_to_lds(g0, g1, gz4, gz4, gz8, 0);
        __builtin_amdgcn_s_wait_tensorcnt((short)0);
    }
    __syncthreads();
    // 4th staged float was feats[:,3]; overwrite with the running-min init.
    for (int i = tid; i < RPB; i += TPB) tile[i].w = 1e10f;
#else
    for (int i = tid; i < RPB; i += TPB) {
        const float* s = feats + (size_t)(base + i) * FWIDTH;
        tile[i] = make_float4(s[0], s[1], s[2], 1e10f);
    }
#endif

    if (tid == 0) {
        qshared[0] = feats[0];
        qshared[1] = feats[1];
        qshared[2] = feats[2];
    }
    __syncthreads();

    for (int step = 0; step < MSAMP - 1; ++step) {
        const float qx = qshared[0], qy = qshared[1], qz = qshared[2];

        // ---- Local pass: pure LDS traffic (ds_load_b128 / ds_store_b32) ----
        unsigned long long best = 0ull;
        for (int i = tid; i < RPB; i += TPB) {
            float4 p = tile[i];
            float dx = p.x - qx, dy = p.y - qy, dz = p.z - qz;
            float d  = fmaf(dz, dz, fmaf(dy, dy, dx * dx));
            float nd = fminf(p.w, d);
            tile[i].w = nd;
            best = u64max(best, pack_vi(nd, (unsigned)(base + i)));
        }

        // ---- Block argmax: wave32 shuffle reduce + LDS combine ----
        for (int off = 16; off > 0; off >>= 1)
            best = u64max(best, __shfl_down(best, (unsigned)off, 32));
        if ((tid & 31) == 0) wred[tid >> 5] = best;
        __syncthreads();

        // ---- Publish + grid barrier (monotonic counter, agent scope) ----
        if (tid == 0) {
            unsigned long long bb = wred[0];
            for (int w = 1; w < TPB / 32; ++w) bb = u64max(bb, wred[w]);
            __hip_atomic_store(&blk_best[b], bb, __ATOMIC_RELEASE,
                               __HIP_MEMORY_SCOPE_AGENT);
            __hip_atomic_fetch_add(counter, 1u, __ATOMIC_ACQ_REL,
                                   __HIP_MEMORY_SCOPE_AGENT);
            const unsigned target = (unsigned)(step + 1) * (unsigned)NBLK;
            while (__hip_atomic_load(counter, __ATOMIC_ACQUIRE,
                                     __HIP_MEMORY_SCOPE_AGENT) < target)
                __builtin_amdgcn_s_sleep(1);
        }
        __syncthreads();  // all threads gated behind thread 0's spin

        // ---- Global argmax over 128 per-block results (parallel L2 reads) --
        unsigned long long g = 0ull;
        if (tid < NBLK)
            g = __hip_atomic_load(&blk_best[tid], __ATOMIC_RELAXED,
                                  __HIP_MEMORY_SCOPE_AGENT);
        for (int off = 16; off > 0; off >>= 1)
            g = u64max(g, __shfl_down(g, (unsigned)off, 32));
        if ((tid & 31) == 0) wred[tid >> 5] = g;
        __syncthreads();

        if (tid == 0) {
            unsigned long long gb = wred[0];
            for (int w = 1; w < NBLK / 32; ++w) gb = u64max(gb, wred[w]);
            const unsigned nidx = 0xFFFFFFFFu - (unsigned)(gb & 0xFFFFFFFFull);
            const float* s = feats + (size_t)nidx * FWIDTH;  // L2-resident
            qshared[0] = s[0];
            qshared[1] = s[1];
            qshared[2] = s[2];
            if (b == 0) out[step + 1] = (int)nidx;
        }
        __syncthreads();
    }
}

extern "C" void kernel_launch(void* const* d_in, const int* in_sizes, int n_in,
                              void* d_out, int out_size, void* d_ws, size_t ws_size,
                              hipStream_t stream) {
    (void)in_sizes; (void)n_in; (void)out_size; (void)ws_size;
    const float* feats = (const float*)d_in[0];
    int* out = (int*)d_out;   // reference output dtype is int32

    unsigned long long* blk_best = (unsigned long long*)d_ws;
    unsigned int* counter =
        (unsigned int*)((char*)d_ws + NBLK * sizeof(unsigned long long));

    fps_init_kernel<<<1, 32, 0, stream>>>(counter, out);
    fps_main_kernel<<<NBLK, TPB, 0, stream>>>(feats, out, blk_best, counter);
}